// GCN_10900626997876
// MI455X (gfx1250) — compile-verified
//
#include <hip/hip_runtime.h>

typedef float v2f __attribute__((ext_vector_type(2)));
typedef float v8f __attribute__((ext_vector_type(8)));

// ---------------------------------------------------------------------------
// init: deg = 1 (self loop), agg1 = 0, agg2 = 0   (grid covers 16N)
// ---------------------------------------------------------------------------
__global__ void k_init(float* __restrict__ deg, float* __restrict__ agg1,
                       float* __restrict__ agg2, int N) {
    int t = blockIdx.x * blockDim.x + threadIdx.x;
    if (t < N) deg[t] = 1.0f;
    if (t < 16 * N) agg1[t] = 0.0f;
    if (t < 2 * N) agg2[t] = 0.0f;
}

// ---------------------------------------------------------------------------
// degree: deg[dst[e]] += 1   (f32 atomics, L2-resident; also warms edge
// indices into the 192MB L2 for the later edge passes)
// ---------------------------------------------------------------------------
__global__ void k_degree(const long long* __restrict__ dst, float* __restrict__ deg,
                         int E) {
    int e = blockIdx.x * blockDim.x + threadIdx.x;
    if (e < E) atomicAdd(&deg[(int)dst[e]], 1.0f);
}

__global__ void k_dinv(const float* __restrict__ deg, float* __restrict__ dinv, int N) {
    int i = blockIdx.x * blockDim.x + threadIdx.x;
    if (i < N) dinv[i] = rsqrtf(deg[i]);   // deg >= 1 always (self-loop)
}

// ---------------------------------------------------------------------------
// GEMM1: h1[N,16] = x[N,3] @ W1[3,16] via V_WMMA_F32_16X16X4_F32 (K padded to 4)
// A (32-bit 16x4): lanes 0-15 -> K={0,1}, lanes 16-31 -> K={2,3}, M = lane%16
// B/C/D: Ncol = lane%16; C VGPR r holds M=r (lo half) / M=r+8 (hi half)
// Full tiles are branch-free (loads unconditional, zero via v_cndmask);
// tail tile (N%16) handled by wave 0 only.
// ---------------------------------------------------------------------------
__global__ void __launch_bounds__(256)
k_gemm1(const float* __restrict__ x, const float* __restrict__ W1,
        float* __restrict__ h1, int N) {
    const int lane = threadIdx.x & 31;
    const int wave = (blockIdx.x * blockDim.x + threadIdx.x) >> 5;
    const int nwav = (gridDim.x * blockDim.x) >> 5;
    const int half = lane >> 4;
    const int l15  = lane & 15;
    const int kb   = half * 2;                     // K base for this half-wave
    const int yoff = (half == 0) ? 1 : 0;          // in-bounds addr for pad lane

    v2f b;                                         // W1 padded 4x16 (row 3 = 0)
    b.x = W1[kb * 16 + l15];
    b.y = (half == 0) ? W1[16 + l15] : 0.0f;       // K=1 (lo) / K=3 pad (hi)

    const int full = N >> 4;
    for (int tile = wave; tile < full; tile += nwav) {
        const int row = (tile << 4) + l15;
        v2f a;
        a.x = x[row * 3 + kb];
        const float vy = x[row * 3 + yoff];        // unconditional, in-bounds
        a.y = (half == 0) ? vy : 0.0f;             // v_cndmask, no branch
        v8f c = {};
        c = __builtin_amdgcn_wmma_f32_16x16x4_f32(false, a, false, b,
                                                  (short)0, c, false, false);
        const int obase = ((tile << 4) + half * 8) * 16 + l15;
#pragma unroll
        for (int r = 0; r < 8; ++r) h1[obase + r * 16] = c[r];
    }

    if ((N & 15) && wave == 0) {                   // tail tile, bounds-checked
        const int tile = full;
        const int row  = (tile << 4) + l15;
        v2f a = {0.0f, 0.0f};
        if (row < N) {
            a.x = x[row * 3 + kb];
            a.y = (half == 0) ? x[row * 3 + 1] : 0.0f;
        }
        v8f c = {};
        c = __builtin_amdgcn_wmma_f32_16x16x4_f32(false, a, false, b,
                                                  (short)0, c, false, false);
#pragma unroll
        for (int r = 0; r < 8; ++r) {
            const int orow = (tile << 4) + r + half * 8;
            if (orow < N) h1[orow * 16 + l15] = c[r];
        }
    }
}

// ---------------------------------------------------------------------------
// edge scatter, layer 1: 16 lanes per edge (one per channel), 32-bit indexing
// agg1[d][c] += h1[s][c] * dinv[s]*dinv[d]
// ---------------------------------------------------------------------------
__global__ void k_edge1(const long long* __restrict__ ei, const float* __restrict__ dinv,
                        const float* __restrict__ h1, float* __restrict__ agg1,
                        int E) {
    const unsigned t = blockIdx.x * blockDim.x + threadIdx.x;
    const int e = (int)(t >> 4);
    if (e >= E) return;
    const int c = (int)t & 15;
    const int s = (int)ei[e];
    const int d = (int)ei[E + e];
    const float nrm = dinv[s] * dinv[d];
    atomicAdd(&agg1[d * 16 + c], h1[s * 16 + c] * nrm);
}

// ---------------------------------------------------------------------------
// post1: z = relu(agg1 + h1 * dinv^2 (self loop) + b1), in-place into h1
// ---------------------------------------------------------------------------
__global__ void k_post1(float* __restrict__ h1, const float* __restrict__ agg1,
                        const float* __restrict__ dinv, const float* __restrict__ b1,
                        int N) {
    int t = blockIdx.x * blockDim.x + threadIdx.x;
    if (t >= 16 * N) return;
    const int i = t >> 4;
    const int c = t & 15;
    const float sn = dinv[i] * dinv[i];
    const float v = agg1[t] + h1[t] * sn + b1[c];
    h1[t] = fmaxf(v, 0.0f);
}

// ---------------------------------------------------------------------------
// GEMM2: h2[N,2] = z[N,16] @ W2[16,2] via 4 chained V_WMMA_F32_16X16X4_F32
// B padded to 16x16 (cols 2..15 zero); only N-cols 0,1 stored
// ---------------------------------------------------------------------------
__global__ void __launch_bounds__(256)
k_gemm2(const float* __restrict__ z, const float* __restrict__ W2,
        float* __restrict__ h2, int N) {
    const int lane = threadIdx.x & 31;
    const int wave = (blockIdx.x * blockDim.x + threadIdx.x) >> 5;
    const int nwav = (gridDim.x * blockDim.x) >> 5;
    const int half = lane >> 4;
    const int l15  = lane & 15;

    v2f bt[4];
#pragma unroll
    for (int t = 0; t < 4; ++t) {
        const int kb = 4 * t + 2 * half;
        bt[t].x = (l15 < 2) ? W2[kb * 2 + l15] : 0.0f;
        bt[t].y = (l15 < 2) ? W2[(kb + 1) * 2 + l15] : 0.0f;
    }

    const int full = N >> 4;
    for (int tile = wave; tile < full; tile += nwav) {
        const int abase = ((tile << 4) + l15) * 16;
        v8f c = {};
#pragma unroll
        for (int t = 0; t < 4; ++t) {
            const int kb = 4 * t + 2 * half;
            v2f a;
            a.x = z[abase + kb];
            a.y = z[abase + kb + 1];
            c = __builtin_amdgcn_wmma_f32_16x16x4_f32(false, a, false, bt[t],
                                                      (short)0, c, false, false);
        }
        if (l15 < 2) {
            const int obase = ((tile << 4) + half * 8) * 2 + l15;
#pragma unroll
            for (int r = 0; r < 8; ++r) h2[obase + r * 2] = c[r];
        }
    }

    if ((N & 15) && wave == 0) {                   // tail tile, bounds-checked
        const int tile = full;
        const int row  = (tile << 4) + l15;
        v8f c = {};
#pragma unroll
        for (int t = 0; t < 4; ++t) {
            const int kb = 4 * t + 2 * half;
            v2f a = {0.0f, 0.0f};
            if (row < N) {
                a.x = z[row * 16 + kb];
                a.y = z[row * 16 + kb + 1];
            }
            c = __builtin_amdgcn_wmma_f32_16x16x4_f32(false, a, false, bt[t],
                                                      (short)0, c, false, false);
        }
        if (l15 < 2) {
#pragma unroll
            for (int r = 0; r < 8; ++r) {
                const int orow = (tile << 4) + r + half * 8;
                if (orow < N) h2[orow * 2 + l15] = c[r];
            }
        }
    }
}

// ---------------------------------------------------------------------------
// edge scatter, layer 2: 2 lanes per edge, 32-bit indexing
// ---------------------------------------------------------------------------
__global__ void k_edge2(const long long* __restrict__ ei, const float* __restrict__ dinv,
                        const float* __restrict__ h2, float* __restrict__ agg2,
                        int E) {
    const unsigned t = blockIdx.x * blockDim.x + threadIdx.x;
    const int e = (int)(t >> 1);
    if (e >= E) return;
    const int c = (int)t & 1;
    const int s = (int)ei[e];
    const int d = (int)ei[E + e];
    const float nrm = dinv[s] * dinv[d];
    atomicAdd(&agg2[d * 2 + c], h2[s * 2 + c] * nrm);
}

// ---------------------------------------------------------------------------
// post2: v = agg2 + h2*dinv^2 + b2 ; out = log_softmax(v) over 2 classes
// ---------------------------------------------------------------------------
__global__ void k_post2(const float* __restrict__ h2, const float* __restrict__ agg2,
                        const float* __restrict__ dinv, const float* __restrict__ b2,
                        float* __restrict__ out, int N) {
    int i = blockIdx.x * blockDim.x + threadIdx.x;
    if (i >= N) return;
    const float sn = dinv[i] * dinv[i];
    const float v0 = agg2[2 * i]     + h2[2 * i]     * sn + b2[0];
    const float v1 = agg2[2 * i + 1] + h2[2 * i + 1] * sn + b2[1];
    const float m  = fmaxf(v0, v1);
    const float lse = m + __logf(__expf(v0 - m) + __expf(v1 - m));
    out[2 * i]     = v0 - lse;
    out[2 * i + 1] = v1 - lse;
}

// ---------------------------------------------------------------------------
extern "C" void kernel_launch(void* const* d_in, const int* in_sizes, int n_in,
                              void* d_out, int out_size, void* d_ws, size_t ws_size,
                              hipStream_t stream) {
    (void)n_in; (void)out_size; (void)ws_size;
    const float*     x  = (const float*)d_in[0];
    const long long* ei = (const long long*)d_in[1];   // int64 [2, E] row-major
    const float*     W1 = (const float*)d_in[2];
    const float*     b1 = (const float*)d_in[3];
    const float*     W2 = (const float*)d_in[4];
    const float*     b2 = (const float*)d_in[5];
    float*           out = (float*)d_out;

    const int N = in_sizes[0] / 3;
    const int E = in_sizes[1] / 2;

    // workspace layout (floats): deg[N] dinv[N] h1[16N] agg1[16N] h2[2N] agg2[2N]
    float* ws   = (float*)d_ws;
    float* deg  = ws;
    float* dinv = ws + (long long)N;
    float* h1   = ws + 2LL  * N;
    float* agg1 = ws + 18LL * N;
    float* h2   = ws + 34LL * N;
    float* agg2 = ws + 36LL * N;

    const int B = 256;
    auto cdiv = [](long long a, long long b) { return (unsigned)((a + b - 1) / b); };

    k_init  <<<cdiv(16LL * N, B), B, 0, stream>>>(deg, agg1, agg2, N);
    k_degree<<<cdiv(E, B),        B, 0, stream>>>(ei + E, deg, E);
    k_dinv  <<<cdiv(N, B),        B, 0, stream>>>(deg, dinv, N);
    k_gemm1 <<<128,               B, 0, stream>>>(x, W1, h1, N);
    k_edge1 <<<cdiv(16LL * E, B), B, 0, stream>>>(ei, dinv, h1, agg1, E);
    k_post1 <<<cdiv(16LL * N, B), B, 0, stream>>>(h1, agg1, dinv, b1, N);
    k_gemm2 <<<128,               B, 0, stream>>>(h1, W2, h2, N);
    k_edge2 <<<cdiv(2LL * E, B),  B, 0, stream>>>(ei, dinv, h2, agg2, E);
    k_post2 <<<cdiv(N, B),        B, 0, stream>>>(h2, agg2, dinv, b2, out, N);
}